// Attn_64364379898032
// MI455X (gfx1250) — compile-verified
//
#include <hip/hip_runtime.h>
#include <cmath>

#define BATCH 64
#define SEQ   4096
#define HID   256
#define NWAVES 8

typedef __attribute__((ext_vector_type(2))) float v2f;
typedef __attribute__((ext_vector_type(4))) float v4f;
typedef __attribute__((ext_vector_type(8))) float v8f;

#if defined(__gfx1250__) && __has_builtin(__builtin_amdgcn_wmma_f32_16x16x4_f32)
#define USE_WMMA_F32 1
#endif

__device__ __forceinline__ float rdlane_f32(float v, int lane) {
  return __builtin_bit_cast(float, __builtin_amdgcn_readlane(__builtin_bit_cast(int, v), lane));
}

// Phase 1: one block per (batch, S-chunk). 8 waves; each wave streams its rows once,
// maintaining an online-softmax state (m, l, ctx[8 cols/lane]) entirely in registers.
// Energies for 16-row tiles computed with V_WMMA_F32_16X16X4_F32 (exact f32 FMA).
__global__ __launch_bounds__(256) void attn_phase1(
    const float* __restrict__ hid,      // (B, H)
    const float* __restrict__ enc,      // (B, S, H)
    float* __restrict__ ctx_part,       // (B*split, H)    unnormalized partial context
    float* __restrict__ ml_part,        // (B*split, 2)    {m, l} per chunk
    float* __restrict__ out,            // (B, H)          used when is_final
    int split, int rows_per_block, int is_final)
{
  __shared__ float lds_h[HID];
  __shared__ float lds_ctx[NWAVES * HID];
  __shared__ float lds_m[NWAVES];
  __shared__ float lds_l[NWAVES];

  const int tid   = threadIdx.x;
  const int wave  = tid >> 5;
  const int lane  = tid & 31;
  const int b     = blockIdx.x / split;
  const int chunk = blockIdx.x % split;

  // Stage h[b] into LDS once (feeds the WMMA B operand every K-step).
  lds_h[tid] = hid[b * HID + tid];
  __syncthreads();

  const int rows_per_wave = rows_per_block / NWAVES;
  const int ntiles        = rows_per_wave >> 4;
  const int row0          = chunk * rows_per_block + wave * rows_per_wave;
  const float* __restrict__ Eb = enc + (size_t)b * SEQ * HID;

  float m = -INFINITY;
  float l = 0.0f;
  float ctx[8];
#pragma unroll
  for (int i = 0; i < 8; ++i) ctx[i] = 0.0f;

  const int mrow = lane & 15;        // matrix row owned by this lane (A layout)
  const int ksub = (lane >> 4) << 1; // lane-half K offset: 0 or 2

  for (int t = 0; t < ntiles; ++t) {
    const int r0 = row0 + (t << 4);

    // Prefetch next tile's rows (one line per row) while this tile computes.
    if (t + 1 < ntiles)
      __builtin_prefetch(Eb + (size_t)(r0 + 16 + mrow) * HID, 0, 0);

    float e[16];

#ifdef USE_WMMA_F32
    // ---- energies for 16 rows via chained f32 WMMA ----
    // A: 16x4 tile of E (lane=row, lane-half holds K-pair {ksub, ksub+1})
    // B: 4x16 broadcast of h chunk (all 16 columns identical)
    const float* __restrict__ arow = Eb + (size_t)(r0 + mrow) * HID + ksub;
    v8f acc = {0.f, 0.f, 0.f, 0.f, 0.f, 0.f, 0.f, 0.f};
#pragma unroll
    for (int k0 = 0; k0 < HID; k0 += 4) {
      v2f a  = *(const v2f*)(arow + k0);            // global_load_b64 (8B/lane)
      v2f bb = *(const v2f*)(&lds_h[k0 + ksub]);    // ds_load_b64
      acc = __builtin_amdgcn_wmma_f32_16x16x4_f32(
          false, a, false, bb, (short)0, acc, false, false);
    }
    // D[m,n] identical across columns n: lane 0 holds rows 0..7, lane 16 rows 8..15.
#pragma unroll
    for (int j = 0; j < 8; ++j) {
      e[j]     = rdlane_f32(acc[j], 0);
      e[j + 8] = rdlane_f32(acc[j], 16);
    }
#else
    // ---- fallback: VALU dot + wave shuffle reduction ----
    {
      const v4f hv0 = *(const v4f*)(&lds_h[lane * 8]);
      const v4f hv1 = *(const v4f*)(&lds_h[lane * 8 + 4]);
#pragma unroll
      for (int j = 0; j < 16; ++j) {
        const float* row = Eb + (size_t)(r0 + j) * HID + lane * 8;
        v4f x0 = *(const v4f*)(row);
        v4f x1 = *(const v4f*)(row + 4);
        float p = x0.x * hv0.x + x0.y * hv0.y + x0.z * hv0.z + x0.w * hv0.w
                + x1.x * hv1.x + x1.y * hv1.y + x1.z * hv1.z + x1.w * hv1.w;
#pragma unroll
        for (int off = 16; off > 0; off >>= 1) p += __shfl_xor(p, off, 32);
        e[j] = p;
      }
    }
#endif

    // ---- wave-uniform online softmax update ----
    float tmax = e[0];
#pragma unroll
    for (int j = 1; j < 16; ++j) tmax = fmaxf(tmax, e[j]);
    const float m_new = fmaxf(m, tmax);
    const float scale = __expf(m - m_new);
    float w[16];
    float lsum = 0.0f;
#pragma unroll
    for (int j = 0; j < 16; ++j) { w[j] = __expf(e[j] - m_new); lsum += w[j]; }
    l = l * scale + lsum;
    m = m_new;
#pragma unroll
    for (int i = 0; i < 8; ++i) ctx[i] *= scale;

    // ---- context accumulation: coalesced b128 reload of the hot tile ----
    const float* __restrict__ rowp = Eb + (size_t)r0 * HID + lane * 8;
#pragma unroll
    for (int j = 0; j < 16; ++j) {
      v4f x0 = *(const v4f*)(rowp);
      v4f x1 = *(const v4f*)(rowp + 4);
      rowp += HID;
      const float wj = w[j];
      ctx[0] = fmaf(wj, x0.x, ctx[0]);
      ctx[1] = fmaf(wj, x0.y, ctx[1]);
      ctx[2] = fmaf(wj, x0.z, ctx[2]);
      ctx[3] = fmaf(wj, x0.w, ctx[3]);
      ctx[4] = fmaf(wj, x1.x, ctx[4]);
      ctx[5] = fmaf(wj, x1.y, ctx[5]);
      ctx[6] = fmaf(wj, x1.z, ctx[6]);
      ctx[7] = fmaf(wj, x1.w, ctx[7]);
    }
  }

  // ---- merge the 8 waves' partials via LDS ----
#pragma unroll
  for (int i = 0; i < 8; ++i) lds_ctx[wave * HID + lane * 8 + i] = ctx[i];
  if (lane == 0) { lds_m[wave] = m; lds_l[wave] = l; }
  __syncthreads();

  float M = lds_m[0];
#pragma unroll
  for (int w2 = 1; w2 < NWAVES; ++w2) M = fmaxf(M, lds_m[w2]);
  float L = 0.0f, cx = 0.0f;
#pragma unroll
  for (int w2 = 0; w2 < NWAVES; ++w2) {
    const float f = __expf(lds_m[w2] - M);
    L += lds_l[w2] * f;
    cx += lds_ctx[w2 * HID + tid] * f;
  }

  if (is_final) {
    out[b * HID + tid] = cx / L;
  } else {
    ctx_part[(size_t)(b * split + chunk) * HID + tid] = cx;
    if (tid == 0) {
      ml_part[(b * split + chunk) * 2 + 0] = M;
      ml_part[(b * split + chunk) * 2 + 1] = L;
    }
  }
}

// Phase 2: log-sum-exp merge across S-chunks. Tiny (B*H threads total).
__global__ __launch_bounds__(256) void attn_phase2(
    const float* __restrict__ ctx_part,
    const float* __restrict__ ml_part,
    float* __restrict__ out, int split)
{
  const int b = blockIdx.x;
  const int t = threadIdx.x;
  float M = -INFINITY;
  for (int c = 0; c < split; ++c)
    M = fmaxf(M, ml_part[(b * split + c) * 2]);
  float L = 0.0f, cx = 0.0f;
  for (int c = 0; c < split; ++c) {
    const float f = __expf(ml_part[(b * split + c) * 2] - M);
    L += ml_part[(b * split + c) * 2 + 1] * f;
    cx += ctx_part[(size_t)(b * split + c) * HID + t] * f;
  }
  out[b * HID + t] = cx / L;
}

extern "C" void kernel_launch(void* const* d_in, const int* in_sizes, int n_in,
                              void* d_out, int out_size, void* d_ws, size_t ws_size,
                              hipStream_t stream) {
  (void)in_sizes; (void)n_in; (void)out_size;
  const float* hid = (const float*)d_in[0];   // (1, B, H) f32
  const float* enc = (const float*)d_in[1];   // (B, S, H) f32
  float* out = (float*)d_out;                 // (B, H) f32

  // Pick the largest S-split whose partials fit in the workspace.
  int split = 1;
  for (int s = 16; s > 1; s >>= 1) {
    const size_t need = (size_t)BATCH * s * (HID + 2) * sizeof(float);
    if (ws_size >= need) { split = s; break; }
  }

  float* ctx_part = (float*)d_ws;
  float* ml_part  = ctx_part + (size_t)BATCH * split * HID;

  dim3 block(256);
  dim3 grid1(BATCH * split);
  attn_phase1<<<grid1, block, 0, stream>>>(hid, enc, ctx_part, ml_part, out,
                                           split, SEQ / split, split == 1 ? 1 : 0);
  if (split > 1) {
    attn_phase2<<<dim3(BATCH), block, 0, stream>>>(ctx_part, ml_part, out, split);
  }
}